// LossLayer_57312043598528
// MI455X (gfx1250) — compile-verified
//
#include <hip/hip_runtime.h>
#include <hip/hip_bf16.h>

// B=4096, R=64, D=512.  dist = sqrt(||w||^2 + ||e||^2 - 2 w.e); GEMM via
// V_WMMA_F32_16X16X4_F32 (native f32 matrix op on CDNA5, wave32).

typedef __attribute__((ext_vector_type(2))) float v2f;
typedef __attribute__((ext_vector_type(8))) float v8f;

#define NB 4096
#define NR 64
#define ND 512

// ---------------------------------------------------------------------------
// Kernel 1: dist[b,r] for a 16-row strip per wave, N=64 in 4 WMMA tiles.
// Block = 128 threads = 4 waves; grid = 4096/(16*4) = 64 blocks.
// dist is written into the pred region of d_out (same 4096x64 f32 footprint).
// ---------------------------------------------------------------------------
__global__ __launch_bounds__(128) void dist_wmma_kernel(
    const float* __restrict__ W,   // [4096,512]
    const float* __restrict__ E,   // [64,512]
    float* __restrict__ dist)      // [4096,64]
{
  const int wave = threadIdx.x >> 5;
  const int lane = threadIdx.x & 31;
  const int half = lane >> 4;      // 0: lanes 0-15, 1: lanes 16-31
  const int mrow = lane & 15;
  const int m0   = (blockIdx.x * 4 + wave) * 16;

  // A layout (32-bit 16x4): lanes 0-15 hold K={0,1}, lanes 16-31 hold K={2,3}
  const float* wrow = W + (size_t)(m0 + mrow) * ND + 2 * half;

  v8f acc[4];
  #pragma unroll
  for (int t = 0; t < 4; ++t) acc[t] = (v8f){0.f,0.f,0.f,0.f,0.f,0.f,0.f,0.f};

  float w2p = 0.f;
  float e2p[4] = {0.f, 0.f, 0.f, 0.f};

  #pragma unroll 4
  for (int k0 = 0; k0 < ND; k0 += 4) {
    const float2 a2 = *(const float2*)(wrow + k0);
    w2p = fmaf(a2.x, a2.x, fmaf(a2.y, a2.y, w2p));
    v2f a; a.x = a2.x; a.y = a2.y;

    #pragma unroll
    for (int t = 0; t < 4; ++t) {
      // B = E^T tile (4x16): lane needs E[n, k0+half] and E[n, k0+2+half]
      const float4 e4 = *(const float4*)(E + (size_t)(t * 16 + mrow) * ND + k0);
      v2f b;
      b.x = half ? e4.y : e4.x;
      b.y = half ? e4.w : e4.z;
      e2p[t] = fmaf(b.x, b.x, fmaf(b.y, b.y, e2p[t]));
      acc[t] = __builtin_amdgcn_wmma_f32_16x16x4_f32(
          /*neg_a=*/false, a, /*neg_b=*/false, b,
          /*c_mod=*/(short)0, acc[t], /*reuse_a=*/false, /*reuse_b=*/false);
    }
  }

  // Combine the two half-wave partial norms (each half covered k mod 4 classes)
  const float w2 = w2p + __shfl_xor(w2p, 16, 32);   // lane L holds row (L&15)
  float e2[4];
  #pragma unroll
  for (int t = 0; t < 4; ++t) e2[t] = e2p[t] + __shfl_xor(e2p[t], 16, 32);

  // C layout: VGPR v, lanes 0-15 -> M=v, lanes 16-31 -> M=v+8; N = lane&15
  #pragma unroll
  for (int t = 0; t < 4; ++t) {
    #pragma unroll
    for (int v = 0; v < 8; ++v) {
      const int m = v + 8 * half;
      const float w2v = __shfl(w2, m, 32);          // row m's norm lives in lane m
      const float s   = acc[t][v];
      float d = w2v + e2[t] - 2.0f * s;
      d = sqrtf(fmaxf(d, 0.0f));
      dist[(size_t)(m0 + m) * NR + t * 16 + (lane & 15)] = d;
    }
  }
}

// ---------------------------------------------------------------------------
// Kernel 2: one wave per row b. Top-2 over dist row (first-index tie-break,
// matching jnp.argmax / lax.top_k), label argmax, margin loss, one-hot pred.
// Reads dist from `dp` then overwrites the same row with pred.
// Block = 256 = 8 waves; grid = 512.
// ---------------------------------------------------------------------------
__global__ __launch_bounds__(256) void row_reduce_kernel(
    float* dp,                       // in: dist [4096,64]; out: pred one-hot
    const float* __restrict__ label, // [4096,64]
    float* __restrict__ loss_ws)     // [4096] per-row losses
{
  const int wave = threadIdx.x >> 5;
  const int lane = threadIdx.x & 31;
  const int b    = blockIdx.x * 8 + wave;

  float* drow = dp + (size_t)b * NR;
  const float d0 = drow[lane];
  const float d1 = drow[lane + 32];

  // local top-2 (index lane < lane+32, so >= keeps first occurrence on tie)
  float t1v, t2v; int t1i, t2i;
  if (d0 >= d1) { t1v = d0; t1i = lane;      t2v = d1; t2i = lane + 32; }
  else          { t1v = d1; t1i = lane + 32; t2v = d0; t2i = lane;      }

  #pragma unroll
  for (int off = 16; off > 0; off >>= 1) {
    const float o1v = __shfl_xor(t1v, off, 32);
    const int   o1i = __shfl_xor(t1i, off, 32);
    const float o2v = __shfl_xor(t2v, off, 32);
    const int   o2i = __shfl_xor(t2i, off, 32);
    const bool o1_beats_t1 = (o1v > t1v) || (o1v == t1v && o1i < t1i);
    if (o1_beats_t1) {
      // #2 of union = better of (old t1, o2)
      float c2v; int c2i;
      if ((t1v > o2v) || (t1v == o2v && t1i < o2i)) { c2v = t1v; c2i = t1i; }
      else                                          { c2v = o2v; c2i = o2i; }
      t1v = o1v; t1i = o1i; t2v = c2v; t2i = c2i;
    } else {
      if ((o1v > t2v) || (o1v == t2v && o1i < t2i)) { t2v = o1v; t2i = o1i; }
    }
  }

  // label argmax (first index on tie)
  const float l0 = label[(size_t)b * NR + lane];
  const float l1 = label[(size_t)b * NR + lane + 32];
  float mv; int mi;
  if (l0 >= l1) { mv = l0; mi = lane; } else { mv = l1; mi = lane + 32; }
  #pragma unroll
  for (int off = 16; off > 0; off >>= 1) {
    const float ov = __shfl_xor(mv, off, 32);
    const int   oi = __shfl_xor(mi, off, 32);
    if ((ov > mv) || (ov == mv && oi < mi)) { mv = ov; mi = oi; }
  }
  const int y = mi;

  const float plus  = (y < 32) ? __shfl(d0, y, 32) : __shfl(d1, y - 32, 32);
  const float minus = (t1i == y) ? t2v : t1v;
  if (lane == 0) loss_ws[b] = 1.0f + plus - minus;

  // one-hot pred (depends on t1i -> ordered after the dist loads above)
  drow[lane]      = (lane      == t1i) ? 1.0f : 0.0f;
  drow[lane + 32] = (lane + 32 == t1i) ? 1.0f : 0.0f;
}

// ---------------------------------------------------------------------------
// Kernel 3: mean of 4096 per-row losses -> out[B*R]
// ---------------------------------------------------------------------------
__global__ __launch_bounds__(256) void loss_mean_kernel(
    const float* __restrict__ loss_ws, float* __restrict__ out)
{
  __shared__ float red[8];
  const int tid = threadIdx.x;
  float s = 0.f;
  for (int i = tid; i < NB; i += 256) s += loss_ws[i];
  #pragma unroll
  for (int off = 16; off > 0; off >>= 1) s += __shfl_xor(s, off, 32);
  if ((tid & 31) == 0) red[tid >> 5] = s;
  __syncthreads();
  if (tid == 0) {
    float tot = 0.f;
    #pragma unroll
    for (int i = 0; i < 8; ++i) tot += red[i];
    out[(size_t)NB * NR] = tot / (float)NB;
  }
}

extern "C" void kernel_launch(void* const* d_in, const int* in_sizes, int n_in,
                              void* d_out, int out_size, void* d_ws, size_t ws_size,
                              hipStream_t stream) {
  const float* W     = (const float*)d_in[0];  // w_output   [4096,512]
  const float* E     = (const float*)d_in[1];  // r_embedding[64,512]
  const float* label = (const float*)d_in[2];  // label      [4096,64]
  float* out     = (float*)d_out;              // pred [4096,64] ++ loss [1]
  float* loss_ws = (float*)d_ws;               // 4096 floats (16 KB)

  // 1) dist into pred region of d_out (same footprint, reused as scratch)
  dist_wmma_kernel<<<dim3(NB / (16 * 4)), dim3(128), 0, stream>>>(W, E, out);
  // 2) per-row top-2 / argmax / loss; overwrites dist rows with one-hot pred
  row_reduce_kernel<<<dim3(NB / 8), dim3(256), 0, stream>>>(out, label, loss_ws);
  // 3) mean loss
  loss_mean_kernel<<<dim3(1), dim3(256), 0, stream>>>(loss_ws, out);
}